// SymplecticGyroceptron_50337016709337
// MI455X (gfx1250) — compile-verified
//
#include <hip/hip_runtime.h>

typedef __attribute__((ext_vector_type(2))) float v2f;
typedef __attribute__((ext_vector_type(8))) float v8f;

#define NIU 64       // hidden units
#define NPSI 8
#define NNI 8
#define EPS_NI 0.01f

__device__ __forceinline__ float tanh_fast(float x) {
#if defined(__has_builtin)
#if __has_builtin(__builtin_amdgcn_tanhf)
  return __builtin_amdgcn_tanhf(x);   // gfx1250 v_tanh_f32
#else
  return tanhf(x);
#endif
#else
  return tanhf(x);
#endif
}

// Per-layer constants held in registers for the 4 map iterations of a layer.
struct LayerC {
  v2f a[4];      // A-matrix tiles: W_in^T (16xK4), lanes16-31 zero (K=2,3 pad)
  v8f bias[4];   // bias in C-layout (VGPR v, lane-half selects M block)
  v8f wout[4];   // W_out in C-layout
  v8f w0[4];     // W_in row 0 in C-layout (for down-projection)
  v8f w1[4];     // W_in row 1 in C-layout
  float eta0, eta1;
};

__device__ __forceinline__ void load_layer(LayerC& L,
                                           const float* __restrict__ Win,   // (2,64)
                                           const float* __restrict__ Wout,  // (64,)
                                           const float* __restrict__ b,     // (64,)
                                           const float* __restrict__ eta,   // (2,)
                                           int n, int half) {
#pragma unroll
  for (int t = 0; t < 4; ++t) {
    float a0 = Win[0 * NIU + 16 * t + n];
    float a1 = Win[1 * NIU + 16 * t + n];
    L.a[t].x = half ? 0.0f : a0;   // lanes 16-31 carry K=2,3 -> zero padding
    L.a[t].y = half ? 0.0f : a1;
#pragma unroll
    for (int v = 0; v < 8; ++v) {
      int m = 16 * t + v + 8 * half;   // C-layout row index
      L.bias[t][v] = b[m];
      L.wout[t][v] = Wout[m];
      L.w0[t][v]   = Win[m];
      L.w1[t][v]   = Win[NIU + m];
    }
  }
  L.eta0 = eta[0];
  L.eta1 = eta[1];
}

// grad_V for two 16-sample tiles at once:
//   t = tanh(Y @ W_in + b);  g = ((1-t^2) * W_out) @ W_in^T
__device__ __forceinline__ void grad_V2(const LayerC& L,
                                        const float y0[2], const float y1[2],
                                        float g0[2], float g1[2]) {
  v2f bm0, bm1;
  bm0.x = y0[0]; bm0.y = y1[0];
  bm1.x = y0[1]; bm1.y = y1[1];
  float s00 = 0.f, s10 = 0.f, s01 = 0.f, s11 = 0.f;
#pragma unroll
  for (int t = 0; t < 4; ++t) {
    // D = A(W_in^T tile) x B(y) + C(bias)  -- exact f32 MACs
    v8f acc0 = __builtin_amdgcn_wmma_f32_16x16x4_f32(
        false, L.a[t], false, bm0, (short)0, L.bias[t], false, false);
    v8f acc1 = __builtin_amdgcn_wmma_f32_16x16x4_f32(
        false, L.a[t], false, bm1, (short)0, L.bias[t], false, false);
#pragma unroll
    for (int v = 0; v < 8; ++v) {
      float t0 = tanh_fast(acc0[v]);
      float t1 = tanh_fast(acc1[v]);
      float u0 = __builtin_fmaf(-t0, t0, 1.0f) * L.wout[t][v];
      float u1 = __builtin_fmaf(-t1, t1, 1.0f) * L.wout[t][v];
      s00 = __builtin_fmaf(u0, L.w0[t][v], s00);
      s10 = __builtin_fmaf(u0, L.w1[t][v], s10);
      s01 = __builtin_fmaf(u1, L.w0[t][v], s01);
      s11 = __builtin_fmaf(u1, L.w1[t][v], s11);
    }
  }
  // merge the two M-halves (lane L and lane L^16 hold complementary m ranges)
  s00 += __shfl_xor(s00, 16, 32);
  s10 += __shfl_xor(s10, 16, 32);
  s01 += __shfl_xor(s01, 16, 32);
  s11 += __shfl_xor(s11, 16, 32);
  g0[0] = s00; g1[0] = s10;
  g0[1] = s01; g1[1] = s11;
}

__device__ __forceinline__ void henon_forward4(const LayerC& L, float eps,
                                               float x0[2], float x1[2],
                                               float y0[2], float y1[2]) {
  for (int it = 0; it < 4; ++it) {
    float g0[2], g1[2];
    grad_V2(L, y0, y1, g0, g1);
#pragma unroll
    for (int j = 0; j < 2; ++j) {
      float nx0 = y0[j] + L.eta0;
      float nx1 = y1[j] + L.eta1;
      float ny0 = __builtin_fmaf(eps, g0[j], -x0[j]);
      float ny1 = __builtin_fmaf(eps, g1[j], -x1[j]);
      x0[j] = nx0; x1[j] = nx1;
      y0[j] = ny0; y1[j] = ny1;
    }
  }
}

__device__ __forceinline__ void henon_inverse4(const LayerC& L, float eps,
                                               float x0[2], float x1[2],
                                               float y0[2], float y1[2]) {
  for (int it = 0; it < 4; ++it) {
    float yn0[2], yn1[2], g0[2], g1[2];
#pragma unroll
    for (int j = 0; j < 2; ++j) {
      yn0[j] = x0[j] - L.eta0;
      yn1[j] = x1[j] - L.eta1;
    }
    grad_V2(L, yn0, yn1, g0, g1);
#pragma unroll
    for (int j = 0; j < 2; ++j) {
      float nx0 = __builtin_fmaf(eps, g0[j], -y0[j]);
      float nx1 = __builtin_fmaf(eps, g1[j], -y1[j]);
      y0[j] = yn0[j]; y1[j] = yn1[j];
      x0[j] = nx0;    x1[j] = nx1;
    }
  }
}

__global__ void __launch_bounds__(256)
gyroceptron_kernel(const float* __restrict__ r, const float* __restrict__ theta0,
                   const float* __restrict__ psi_Win, const float* __restrict__ psi_Wout,
                   const float* __restrict__ psi_b, const float* __restrict__ psi_eta,
                   const float* __restrict__ ni_Win, const float* __restrict__ ni_Wout,
                   const float* __restrict__ ni_b, const float* __restrict__ ni_eta,
                   float* __restrict__ out, int B) {
  const int lane = threadIdx.x & 31;
  const int wave = threadIdx.x >> 5;
  const int n = lane & 15;     // sample-within-tile == WMMA column
  const int half = lane >> 4;  // M-half selector in C layout, K-half in A/B
  const int base = (blockIdx.x * (blockDim.x >> 5) + wave) * 32;  // 2 tiles/wave

  float x0[2], x1[2], y0[2], y1[2];
  int sidx[2];
#pragma unroll
  for (int j = 0; j < 2; ++j) {
    int s = base + j * 16 + n;
    sidx[j] = s;
    int sc = s < B ? s : (B - 1);  // clamp: keep EXEC all-ones for WMMA
    x0[j] = r[4 * sc + 0];  // q1
    x1[j] = r[4 * sc + 1];  // q2
    y0[j] = r[4 * sc + 2];  // p1
    y1[j] = r[4 * sc + 3];  // p2
  }

  LayerC L;

  // inverse psi layers, i = 7..0, eps = 1
  for (int i = NPSI - 1; i >= 0; --i) {
    load_layer(L, psi_Win + i * 2 * NIU, psi_Wout + i * NIU,
               psi_b + i * NIU, psi_eta + i * 2, n, half);
    henon_inverse4(L, 1.0f, x0, x1, y0, y1);
  }

  // circle action on (q1, p1)
  {
    float th = theta0[0];
    float c = cosf(th), s = sinf(th);
#pragma unroll
    for (int j = 0; j < 2; ++j) {
      float q1 = x0[j], p1 = y0[j];
      x0[j] = c * q1 + s * p1;
      y0[j] = c * p1 - s * q1;
    }
  }

  // forward psi layers, i = 0..7, eps = 1
  for (int i = 0; i < NPSI; ++i) {
    load_layer(L, psi_Win + i * 2 * NIU, psi_Wout + i * NIU,
               psi_b + i * NIU, psi_eta + i * 2, n, half);
    henon_forward4(L, 1.0f, x0, x1, y0, y1);
  }

  // forward ni layers, i = 0..7, eps = 0.01
  for (int i = 0; i < NNI; ++i) {
    load_layer(L, ni_Win + i * 2 * NIU, ni_Wout + i * NIU,
               ni_b + i * NIU, ni_eta + i * 2, n, half);
    henon_forward4(L, EPS_NI, x0, x1, y0, y1);
  }

  // store (both lane-halves hold identical results; write from half 0 only)
  if (half == 0) {
#pragma unroll
    for (int j = 0; j < 2; ++j) {
      if (sidx[j] < B) {
        out[4 * sidx[j] + 0] = x0[j];
        out[4 * sidx[j] + 1] = x1[j];
        out[4 * sidx[j] + 2] = y0[j];
        out[4 * sidx[j] + 3] = y1[j];
      }
    }
  }
}

extern "C" void kernel_launch(void* const* d_in, const int* in_sizes, int n_in,
                              void* d_out, int out_size, void* d_ws, size_t ws_size,
                              hipStream_t stream) {
  const float* r        = (const float*)d_in[0];
  const float* theta0   = (const float*)d_in[1];
  const float* psi_Win  = (const float*)d_in[2];
  const float* psi_Wout = (const float*)d_in[3];
  const float* psi_b    = (const float*)d_in[4];
  const float* psi_eta  = (const float*)d_in[5];
  const float* ni_Win   = (const float*)d_in[6];
  const float* ni_Wout  = (const float*)d_in[7];
  const float* ni_b     = (const float*)d_in[8];
  const float* ni_eta   = (const float*)d_in[9];
  float* out = (float*)d_out;

  int B = in_sizes[0] / 4;
  const int threads = 256;                 // 8 waves/block
  const int samplesPerBlock = 8 * 32;      // 2 tiles of 16 per wave
  int grid = (B + samplesPerBlock - 1) / samplesPerBlock;
  gyroceptron_kernel<<<grid, threads, 0, stream>>>(
      r, theta0, psi_Win, psi_Wout, psi_b, psi_eta,
      ni_Win, ni_Wout, ni_b, ni_eta, out, B);
}